// LowrankLearnableHash_76570676953615
// MI455X (gfx1250) — compile-verified
//
#include <hip/hip_runtime.h>

#define BLOCK 256
#define C_PLANE 12
#define RESO 256
#define FRESO 64
#define FDIM 32

typedef __attribute__((address_space(1))) int  gint_t;
typedef __attribute__((address_space(3))) int  lint_t;

// ---------------------------------------------------------------------------
// Plane layout transform: [12,256,256] (x3 planes) -> [plane][y][x][12]
// so one bilinear corner = 48 contiguous bytes (3x float4).
// ---------------------------------------------------------------------------
__global__ void transpose_planes_kernel(const float* __restrict__ p01,
                                        const float* __restrict__ p02,
                                        const float* __restrict__ p12,
                                        float* __restrict__ outT) {
  const int per_plane = C_PLANE * RESO * RESO;
  int t = blockIdx.x * blockDim.x + threadIdx.x;
  if (t >= 3 * per_plane) return;
  int plane = t / per_plane;
  int rem   = t - plane * per_plane;
  int ch    = rem / (RESO * RESO);
  int pix   = rem - ch * (RESO * RESO);
  const float* src = (plane == 0) ? p01 : ((plane == 1) ? p02 : p12);
  float v = src[ch * (RESO * RESO) + pix];
  outT[(size_t)(plane * (RESO * RESO) + pix) * C_PLANE + ch] = v;
}

// ---------------------------------------------------------------------------
// Feature layout transform: [32, 64^3] -> [64^3, 32] via 32x32 LDS tiles so
// one trilinear corner (32 channels) = one 128B cache line.
// ---------------------------------------------------------------------------
__global__ void __launch_bounds__(256)
transpose_feat_kernel(const float* __restrict__ in, float* __restrict__ outT) {
  __shared__ float tile[32][33];
  const int V = FRESO * FRESO * FRESO;  // 262144
  int v0 = blockIdx.x * 32;
  int tx = threadIdx.x & 31;
  int ty = threadIdx.x >> 5;            // 8 row-groups
  #pragma unroll
  for (int c = ty; c < 32; c += 8)
    tile[c][tx] = in[(size_t)c * V + v0 + tx];      // coalesced 128B reads
  __syncthreads();
  #pragma unroll
  for (int r = ty; r < 32; r += 8)
    outT[(size_t)(v0 + r) * FDIM + tx] = tile[tx][r]; // coalesced 128B writes
}

// ---------------------------------------------------------------------------
// Bilinear sample of a 12-channel transposed plane (align_corners, border clamp)
// ---------------------------------------------------------------------------
__device__ __forceinline__ void bilerp_plane(const float* __restrict__ pl,
                                             float cx, float cy,
                                             float acc[C_PLANE], bool first) {
  float x = fminf(fmaxf((cx + 1.0f) * 0.5f * (float)(RESO - 1), 0.0f), (float)(RESO - 1));
  float y = fminf(fmaxf((cy + 1.0f) * 0.5f * (float)(RESO - 1), 0.0f), (float)(RESO - 1));
  float xf = fminf(floorf(x), (float)(RESO - 2));
  float yf = fminf(floorf(y), (float)(RESO - 2));
  int x0 = (int)xf, y0 = (int)yf;
  float wx = x - xf, wy = y - yf;
  float w00 = (1.0f - wx) * (1.0f - wy);
  float w01 = wx * (1.0f - wy);
  float w10 = (1.0f - wx) * wy;
  float w11 = wx * wy;
  const float* base = pl + (size_t)(y0 * RESO + x0) * C_PLANE;
  const float4* r00 = (const float4*)(base);
  const float4* r01 = (const float4*)(base + C_PLANE);
  const float4* r10 = (const float4*)(base + RESO * C_PLANE);
  const float4* r11 = (const float4*)(base + RESO * C_PLANE + C_PLANE);
  #pragma unroll
  for (int i = 0; i < 3; ++i) {
    float4 a = r00[i], b = r01[i], c = r10[i], d = r11[i];
    float v0 = a.x * w00 + b.x * w01 + c.x * w10 + d.x * w11;
    float v1 = a.y * w00 + b.y * w01 + c.y * w10 + d.y * w11;
    float v2 = a.z * w00 + b.z * w01 + c.z * w10 + d.z * w11;
    float v3 = a.w * w00 + b.w * w01 + c.w * w10 + d.w * w11;
    if (first) {
      acc[4 * i + 0] = v0; acc[4 * i + 1] = v1;
      acc[4 * i + 2] = v2; acc[4 * i + 3] = v3;
    } else {
      acc[4 * i + 0] *= v0; acc[4 * i + 1] *= v1;
      acc[4 * i + 2] *= v2; acc[4 * i + 3] *= v3;
    }
  }
}

// ---------------------------------------------------------------------------
// Main kernel: 256 threads = 8 wave32s.
// Phase 0: async-stage pts into LDS (CDNA5 GLOBAL_LOAD_ASYNC_TO_LDS + ASYNCcnt)
// Phase 1: 1 lane = 1 point -> plane products -> coords -> trilinear setup (LDS)
// Phase 2: 1 lane = 1 channel; wave loops its 32 points, fully coalesced
//          128B corner gathers and output stores.
// ---------------------------------------------------------------------------
__global__ void __launch_bounds__(256)
lrh_main_kernel(const float* __restrict__ pts,
                const float* __restrict__ planesT,
                const float* __restrict__ featT,
                float* __restrict__ out, int n) {
  __shared__ float spts[BLOCK * 3];
  __shared__ int   s_base[BLOCK];
  __shared__ float s_wx[BLOCK];
  __shared__ float s_wy[BLOCK];
  __shared__ float s_wz[BLOCK];

  const int tid       = threadIdx.x;
  const int blockBase = blockIdx.x * BLOCK;
  const int elemBase  = blockBase * 3;
  const int totalElem = n * 3;

  // ---- Phase 0: stage point coords (async-to-LDS on gfx1250) ----
#if defined(__AMDGCN__) && __has_builtin(__builtin_amdgcn_global_load_async_to_lds_b32)
  #pragma unroll
  for (int k = 0; k < 3; ++k) {
    int e = tid + k * BLOCK;
    if (elemBase + e < totalElem) {
      __builtin_amdgcn_global_load_async_to_lds_b32(
          (gint_t*)(pts + elemBase + e),
          (lint_t*)(&spts[e]),
          0, 0);
    }
  }
  #if __has_builtin(__builtin_amdgcn_s_wait_asynccnt)
  __builtin_amdgcn_s_wait_asynccnt(0);
  #endif
#else
  #pragma unroll
  for (int k = 0; k < 3; ++k) {
    int e = tid + k * BLOCK;
    if (elemBase + e < totalElem) spts[e] = pts[elemBase + e];
  }
#endif
  __syncthreads();

  // ---- Phase 1: one lane per point ----
  int pid = blockBase + tid;
  if (pid < n) {
    float px = spts[tid * 3 + 0];
    float py = spts[tid * 3 + 1];
    float pz = spts[tid * 3 + 2];

    float acc[C_PLANE];
    const size_t pstride = (size_t)RESO * RESO * C_PLANE;
    bilerp_plane(planesT + 0 * pstride, px, py, acc, true);
    bilerp_plane(planesT + 1 * pstride, px, pz, acc, false);
    bilerp_plane(planesT + 2 * pstride, py, pz, acc, false);

    float cx = acc[0] + acc[1] + acc[2]  + acc[3];
    float cy = acc[4] + acc[5] + acc[6]  + acc[7];
    float cz = acc[8] + acc[9] + acc[10] + acc[11];

    float x = fminf(fmaxf((cx + 1.0f) * 0.5f * (float)(FRESO - 1), 0.0f), (float)(FRESO - 1));
    float y = fminf(fmaxf((cy + 1.0f) * 0.5f * (float)(FRESO - 1), 0.0f), (float)(FRESO - 1));
    float z = fminf(fmaxf((cz + 1.0f) * 0.5f * (float)(FRESO - 1), 0.0f), (float)(FRESO - 1));
    float xf = fminf(floorf(x), (float)(FRESO - 2));
    float yf = fminf(floorf(y), (float)(FRESO - 2));
    float zf = fminf(floorf(z), (float)(FRESO - 2));
    int x0 = (int)xf, y0 = (int)yf, z0 = (int)zf;

    s_base[tid] = ((z0 * FRESO + y0) * FRESO + x0) * FDIM;
    s_wx[tid]   = x - xf;
    s_wy[tid]   = y - yf;
    s_wz[tid]   = z - zf;
  }
  __syncthreads();

  // ---- Phase 2: one lane per channel, wave loops its 32 points ----
  const int wave  = tid >> 5;
  const int c     = tid & 31;
  const int slot0 = wave << 5;
  const int SX = FDIM;                    // +x corner stride
  const int SY = FRESO * FDIM;            // +y corner stride
  const int SZ = FRESO * FRESO * FDIM;    // +z corner stride

  #pragma unroll 1
  for (int j = 0; j < 32; ++j) {
    int p = blockBase + slot0 + j;
    if (p >= n) break;                    // uniform across the wave
    int   b  = s_base[slot0 + j];         // LDS broadcast reads
    float wx = s_wx[slot0 + j];
    float wy = s_wy[slot0 + j];
    float wz = s_wz[slot0 + j];

    const float* f = featT + b + c;       // 8 coalesced 128B gathers
    float v000 = f[0],       v001 = f[SX];
    float v010 = f[SY],      v011 = f[SY + SX];
    float v100 = f[SZ],      v101 = f[SZ + SX];
    float v110 = f[SZ + SY], v111 = f[SZ + SY + SX];

    float fx0 = 1.0f - wx, fy0 = 1.0f - wy, fz0 = 1.0f - wz;
    float r = fz0 * (fy0 * (v000 * fx0 + v001 * wx) +
                     wy  * (v010 * fx0 + v011 * wx)) +
              wz  * (fy0 * (v100 * fx0 + v101 * wx) +
                     wy  * (v110 * fx0 + v111 * wx));

    out[(size_t)p * FDIM + c] = r;        // coalesced 128B store
  }
}

// ---------------------------------------------------------------------------
extern "C" void kernel_launch(void* const* d_in, const int* in_sizes, int n_in,
                              void* d_out, int out_size, void* d_ws, size_t ws_size,
                              hipStream_t stream) {
  const float* pts  = (const float*)d_in[0];
  const float* p01  = (const float*)d_in[1];
  const float* p02  = (const float*)d_in[2];
  const float* p12  = (const float*)d_in[3];
  const float* feat = (const float*)d_in[4];
  int n = in_sizes[0] / 3;

  // workspace layout: [featT 64^3*32 f32 = 32MB][planesT 3*256^2*12 f32 = 9MB]
  float* featT   = (float*)d_ws;
  float* planesT = (float*)((char*)d_ws +
                   (size_t)FRESO * FRESO * FRESO * FDIM * sizeof(float));

  const int V = FRESO * FRESO * FRESO;
  transpose_feat_kernel<<<V / 32, 256, 0, stream>>>(feat, featT);

  const int plane_elems = 3 * C_PLANE * RESO * RESO;
  transpose_planes_kernel<<<(plane_elems + 255) / 256, 256, 0, stream>>>(
      p01, p02, p12, planesT);

  lrh_main_kernel<<<(n + BLOCK - 1) / BLOCK, BLOCK, 0, stream>>>(
      pts, planesT, featT, (float*)d_out, n);
}